// FourScanBranch_78030965834177
// MI455X (gfx1250) — compile-verified
//
#include <hip/hip_runtime.h>
#include <hip/hip_bf16.h>
#include <math.h>

typedef __attribute__((ext_vector_type(2))) float v2f;
typedef __attribute__((ext_vector_type(8))) float v8f;

#define L_   18432   // NF*H*W
#define NG   4       // scan branches
#define DI_  32
#define DS_  16
#define DG_  16
#define TCH  128     // chunk length (timesteps)
#define NCH  144     // L_/TCH

__device__ __forceinline__ float silu_f(float v) {
  return v / (1.0f + __expf(-v));
}
__device__ __forceinline__ float softplus_f(float v) {
  return v > 20.0f ? v : log1pf(__expf(v));
}

// ---------------------------------------------------------------------------
// K1: g = silu(x[perm]*w_in + b_in)  (16 chans), then xz = g @ Win (16x64)
// via chained V_WMMA_F32_16X16X4_F32 (K=16 -> 4 chunks). One wave per N-tile.
// Wave-uniform xc/z split (waves 0-1 -> xc, 2-3 -> z): no exec divergence.
// ---------------------------------------------------------------------------
__global__ __launch_bounds__(128)
void fsb_front_wmma(const float* __restrict__ x, const int* __restrict__ perms,
                    const float* __restrict__ w_in, const float* __restrict__ b_in,
                    const float* __restrict__ Win,
                    float* __restrict__ xcB, float* __restrict__ zB) {
  const int g  = blockIdx.y;
  const int t0 = blockIdx.x * 16;
  const int tid = threadIdx.x;

  __shared__ float gt[16 * 17];  // 16 t-rows x 16 chans, padded

  for (int idx = tid; idx < 256; idx += 128) {
    int t = idx >> 4, c = idx & 15;
    int p = perms[g * L_ + t0 + t];
    float s = x[p];
    float v = fmaf(s, w_in[g * 16 + c], b_in[g * 16 + c]);
    gt[t * 17 + c] = silu_f(v);
  }
  __syncthreads();

  const int lane = tid & 31;
  const int wave = tid >> 5;          // N-tile 0..3
  const int m    = lane & 15;
  const int koff = (lane >> 4) << 1;  // 0 or 2
  const int nb   = wave * 16 + m;

  v8f acc = {0.f, 0.f, 0.f, 0.f, 0.f, 0.f, 0.f, 0.f};
#pragma unroll
  for (int kc = 0; kc < 4; ++kc) {
    int k0 = 4 * kc + koff;
    v2f a; a.x = gt[m * 17 + k0];              a.y = gt[m * 17 + k0 + 1];
    v2f b; b.x = Win[(g * 16 + k0) * 64 + nb]; b.y = Win[(g * 16 + k0 + 1) * 64 + nb];
    acc = __builtin_amdgcn_wmma_f32_16x16x4_f32(false, a, false, b,
                                                (short)0, acc, false, false);
  }

  // wave-uniform destination select; per-lane column within the 32-wide row
  float* __restrict__ dst = (wave < 2) ? xcB : zB;
  const int col = nb & 31;
  const int trow = t0 + ((lane >> 4) << 3);
#pragma unroll
  for (int r = 0; r < 8; ++r) {
    dst[((size_t)g * L_ + trow + r) * 32 + col] = acc[r];
  }
}

// ---------------------------------------------------------------------------
// K2: causal depthwise conv (4 taps) + silu -> u; dbc = u @ Wx (32x33);
// dt = softplus(dbc0*Wdt + bdt); split B,C. One thread per (g,t).
// Tap-outer loop so each conv row vectorizes into b128 loads.
// ---------------------------------------------------------------------------
__global__ __launch_bounds__(256)
void fsb_midmix(const float* __restrict__ xcB,
                const float* __restrict__ convw, const float* __restrict__ convb,
                const float* __restrict__ Wx, const float* __restrict__ Wdt,
                const float* __restrict__ bdt,
                float* __restrict__ uB, float* __restrict__ dtB,
                float* __restrict__ BcB, float* __restrict__ CcB) {
  const int g = blockIdx.y;
  const int tid = threadIdx.x;
  const int t = blockIdx.x * 256 + tid;

  __shared__ float sWx[32 * 33];
  __shared__ float scw[32 * 4];
  __shared__ float scb[32], sWdt[32], sbdt[32];

  for (int idx = tid; idx < 32 * 33; idx += 256) sWx[idx] = Wx[g * 32 * 33 + idx];
  for (int idx = tid; idx < 128; idx += 256)     scw[idx] = convw[g * 128 + idx];
  if (tid < 32) { scb[tid] = convb[g*32+tid]; sWdt[tid] = Wdt[g*32+tid]; sbdt[tid] = bdt[g*32+tid]; }
  __syncthreads();

  float uv[32];
#pragma unroll
  for (int c = 0; c < 32; ++c) uv[c] = scb[c];

#pragma unroll
  for (int k = 0; k < 4; ++k) {
    int tp = t - 3 + k;
    if (tp >= 0) {
      const float* __restrict__ row = xcB + ((size_t)g * L_ + tp) * 32;
#pragma unroll
      for (int c = 0; c < 32; ++c) uv[c] = fmaf(row[c], scw[c * 4 + k], uv[c]);
    }
  }
#pragma unroll
  for (int c = 0; c < 32; ++c) uv[c] = silu_f(uv[c]);

  float dbc0 = 0.f;
  for (int j = 0; j < 33; ++j) {
    float acc = 0.f;
#pragma unroll
    for (int c = 0; c < 32; ++c) acc = fmaf(uv[c], sWx[c * 33 + j], acc);
    if (j == 0)       dbc0 = acc;
    else if (j < 17)  BcB[((size_t)g * L_ + t) * 16 + (j - 1)]  = acc;
    else              CcB[((size_t)g * L_ + t) * 16 + (j - 17)] = acc;
  }

#pragma unroll
  for (int c = 0; c < 32; ++c) {
    uB [((size_t)g * L_ + t) * 32 + c] = uv[c];
    dtB[((size_t)g * L_ + t) * 32 + c] = softplus_f(fmaf(dbc0, sWdt[c], sbdt[c]));
  }
}

// ---------------------------------------------------------------------------
// K3: scan phase 1 — per (g,chunk): local recurrence from h=0 -> end state,
// plus per-channel sum of dt (chunk decay = exp(A*sum_dt)).
// Thread i = d*16+s handles one of 512 state elements. Next-round tiles are
// prefetched (global_prefetch_b8) before the 32-step serial compute.
// ---------------------------------------------------------------------------
__global__ __launch_bounds__(512)
void fsb_scan1(const float* __restrict__ dtB, const float* __restrict__ uB,
               const float* __restrict__ BcB, const float* __restrict__ Alog,
               float* __restrict__ cB, float* __restrict__ cD) {
  const int g = blockIdx.y, ch = blockIdx.x;
  const int i = threadIdx.x, d = i >> 4, s = i & 15;
  const float A = -__expf(Alog[(g * 32 + d) * 16 + s]);

  __shared__ float sdt[32 * 32], su[32 * 32], sB2[32 * 16];

  float h = 0.f, dsum = 0.f;
  const int t0c = ch * TCH;
  for (int r = 0; r < TCH / 32; ++r) {
    const int tb = t0c + r * 32;
    for (int idx = i; idx < 1024; idx += 512) {
      int tt = idx >> 5, c = idx & 31;
      size_t o = ((size_t)g * L_ + tb + tt) * 32 + c;
      sdt[idx] = dtB[o]; su[idx] = uB[o];
    }
    { int tt = i >> 4, c = i & 15;
      sB2[i] = BcB[((size_t)g * L_ + tb + tt) * 16 + c]; }
    __syncthreads();
    if (r < TCH / 32 - 1) {  // hide next tile's latency behind the serial chain
      size_t on = ((size_t)g * L_ + tb + 32 + (i >> 5)) * 32 + (i & 31);
      __builtin_prefetch(dtB + on, 0, 0);
      __builtin_prefetch(uB + on, 0, 0);
    }
#pragma unroll 8
    for (int tt = 0; tt < 32; ++tt) {
      float dtv = sdt[tt * 32 + d];
      float ga  = __expf(dtv * A);
      h = fmaf(ga, h, dtv * sB2[tt * 16 + s] * su[tt * 32 + d]);
      dsum += dtv;
    }
    __syncthreads();
  }
  cB[((size_t)g * NCH + ch) * 512 + i] = h;
  if (s == 0) cD[((size_t)g * NCH + ch) * 32 + d] = dsum;
}

// ---------------------------------------------------------------------------
// K4: scan phase 2 — serial carry over 144 chunks, one block per group.
// ---------------------------------------------------------------------------
__global__ __launch_bounds__(512)
void fsb_scan2(const float* __restrict__ cB, const float* __restrict__ cD,
               const float* __restrict__ Alog, float* __restrict__ Hin) {
  const int g = blockIdx.x;
  const int i = threadIdx.x, d = i >> 4;
  const float A = -__expf(Alog[g * 512 + i]);
  float h = 0.f;
  for (int ch = 0; ch < NCH; ++ch) {
    size_t o = ((size_t)g * NCH + ch) * 512 + i;
    Hin[o] = h;
    h = fmaf(__expf(A * cD[((size_t)g * NCH + ch) * 32 + d]), h, cB[o]);
  }
}

// ---------------------------------------------------------------------------
// K5: scan phase 3 — replay chunk from incoming state, contract with C
// (16-lane butterfly reduce), gate with silu(z), emit yz (L x 32 per group).
// ---------------------------------------------------------------------------
__global__ __launch_bounds__(512)
void fsb_scan3(const float* __restrict__ dtB, const float* __restrict__ uB,
               const float* __restrict__ BcB, const float* __restrict__ CcB,
               const float* __restrict__ zB, const float* __restrict__ Hin,
               const float* __restrict__ Alog, const float* __restrict__ Dp,
               float* __restrict__ yzB) {
  const int g = blockIdx.y, ch = blockIdx.x;
  const int i = threadIdx.x, d = i >> 4, s = i & 15;
  const float A   = -__expf(Alog[(g * 32 + d) * 16 + s]);
  const float Dpv = Dp[g * 32 + d];

  __shared__ float sdt[32 * 32], su[32 * 32], sz[32 * 32];
  __shared__ float sB2[32 * 16], sC2[32 * 16];

  float h = Hin[((size_t)g * NCH + ch) * 512 + i];
  const int t0c = ch * TCH;
  for (int r = 0; r < TCH / 32; ++r) {
    const int tb = t0c + r * 32;
    for (int idx = i; idx < 1024; idx += 512) {
      int tt = idx >> 5, c = idx & 31;
      size_t o = ((size_t)g * L_ + tb + tt) * 32 + c;
      sdt[idx] = dtB[o]; su[idx] = uB[o]; sz[idx] = zB[o];
    }
    { int tt = i >> 4, c = i & 15;
      size_t o = ((size_t)g * L_ + tb + tt) * 16 + c;
      sB2[i] = BcB[o]; sC2[i] = CcB[o]; }
    __syncthreads();
    if (r < TCH / 32 - 1) {
      size_t on = ((size_t)g * L_ + tb + 32 + (i >> 5)) * 32 + (i & 31);
      __builtin_prefetch(dtB + on, 0, 0);
      __builtin_prefetch(uB + on, 0, 0);
      __builtin_prefetch(zB + on, 0, 0);
    }
    for (int tt = 0; tt < 32; ++tt) {
      float dtv = sdt[tt * 32 + d];
      float ga  = __expf(dtv * A);
      h = fmaf(ga, h, dtv * sB2[tt * 16 + s] * su[tt * 32 + d]);
      float c0 = h * sC2[tt * 16 + s];
      c0 += __shfl_xor(c0, 1); c0 += __shfl_xor(c0, 2);
      c0 += __shfl_xor(c0, 4); c0 += __shfl_xor(c0, 8);
      if (s == 0) {
        float uvv = su[tt * 32 + d];
        float yv  = fmaf(Dpv, uvv, c0);
        yzB[((size_t)g * L_ + tb + tt) * 32 + d] = yv * silu_f(sz[tt * 32 + d]);
      }
    }
    __syncthreads();
  }
}

// ---------------------------------------------------------------------------
// K6: proc = yz @ Wout (K=32 -> 8 chained f32 WMMAs) fused with LayerNorm(16),
// per-channel dot w_out and bias -> paths[g][t]. One wave per 16-row tile.
// ---------------------------------------------------------------------------
__global__ __launch_bounds__(128)
void fsb_back_wmma(const float* __restrict__ yzB, const float* __restrict__ Wout,
                   const float* __restrict__ ln_g, const float* __restrict__ ln_b,
                   const float* __restrict__ w_out, const float* __restrict__ b_out,
                   float* __restrict__ paths) {
  const int g    = blockIdx.y;
  const int lane = threadIdx.x & 31;
  const int wave = threadIdx.x >> 5;
  const int t0   = (blockIdx.x * 4 + wave) * 16;
  const int m    = lane & 15;
  const int koff = (lane >> 4) << 1;

  v8f acc = {0.f, 0.f, 0.f, 0.f, 0.f, 0.f, 0.f, 0.f};
  const float* arow = yzB + ((size_t)g * L_ + t0 + m) * 32;
#pragma unroll
  for (int kc = 0; kc < 8; ++kc) {
    int k0 = 4 * kc + koff;
    v2f a; a.x = arow[k0];                      a.y = arow[k0 + 1];
    v2f b; b.x = Wout[(g * 32 + k0) * 16 + m];  b.y = Wout[(g * 32 + k0 + 1) * 16 + m];
    acc = __builtin_amdgcn_wmma_f32_16x16x4_f32(false, a, false, b,
                                                (short)0, acc, false, false);
  }

  const float lg = ln_g[g * 16 + m], lb = ln_b[g * 16 + m];
  const float wo = w_out[g * 16 + m], bo = b_out[g];
#pragma unroll
  for (int r = 0; r < 8; ++r) {
    float v  = acc[r];
    float sm = v, sq = v * v;
    sm += __shfl_xor(sm, 1); sq += __shfl_xor(sq, 1);
    sm += __shfl_xor(sm, 2); sq += __shfl_xor(sq, 2);
    sm += __shfl_xor(sm, 4); sq += __shfl_xor(sq, 4);
    sm += __shfl_xor(sm, 8); sq += __shfl_xor(sq, 8);
    float mean = sm * (1.0f / 16.0f);
    float var  = sq * (1.0f / 16.0f) - mean * mean;
    float rstd = rsqrtf(var + 1e-5f);
    float nm   = fmaf((v - mean) * rstd, lg, lb);
    float pc   = nm * wo;
    pc += __shfl_xor(pc, 1); pc += __shfl_xor(pc, 2);
    pc += __shfl_xor(pc, 4); pc += __shfl_xor(pc, 8);
    if (m == 0) paths[(size_t)g * L_ + t0 + r + ((lane >> 4) << 3)] = pc + bo;
  }
}

// ---------------------------------------------------------------------------
// K7: recons + mean over 4 branches: out[j] = 0.25 * sum_g paths[g][perm[g][j]]
// ---------------------------------------------------------------------------
__global__ __launch_bounds__(256)
void fsb_gather(const float* __restrict__ paths, const int* __restrict__ perms,
                float* __restrict__ out) {
  int j = blockIdx.x * 256 + threadIdx.x;
  if (j >= L_) return;
  float a = 0.f;
#pragma unroll
  for (int g = 0; g < NG; ++g) a += paths[(size_t)g * L_ + perms[g * L_ + j]];
  out[j] = a * 0.25f;
}

// ---------------------------------------------------------------------------
extern "C" void kernel_launch(void* const* d_in, const int* in_sizes, int n_in,
                              void* d_out, int out_size, void* d_ws, size_t ws_size,
                              hipStream_t stream) {
  const float* x      = (const float*)d_in[0];
  const float* w_in   = (const float*)d_in[1];
  const float* b_in   = (const float*)d_in[2];
  const float* w_out  = (const float*)d_in[3];
  const float* b_out  = (const float*)d_in[4];
  const float* Win    = (const float*)d_in[5];
  const float* convw  = (const float*)d_in[6];
  const float* convb  = (const float*)d_in[7];
  const float* Wx     = (const float*)d_in[8];
  const float* Wdt    = (const float*)d_in[9];
  const float* bdt    = (const float*)d_in[10];
  const float* Alog   = (const float*)d_in[11];
  const float* Dp     = (const float*)d_in[12];
  const float* Wout   = (const float*)d_in[13];
  const float* ln_g   = (const float*)d_in[14];
  const float* ln_b   = (const float*)d_in[15];
  const int*   perms  = (const int*)d_in[16];
  float* out = (float*)d_out;

  float* ws   = (float*)d_ws;
  float* xcB  = ws;                          // 4*L*32
  float* zB   = xcB  + (size_t)NG * L_ * 32;
  float* uB   = zB   + (size_t)NG * L_ * 32;
  float* dtB  = uB   + (size_t)NG * L_ * 32;
  float* BcB  = dtB  + (size_t)NG * L_ * 32;
  float* CcB  = BcB  + (size_t)NG * L_ * 16;
  float* yzB  = CcB  + (size_t)NG * L_ * 16;
  float* cBuf = yzB  + (size_t)NG * L_ * 32;  // 4*NCH*512
  float* cDee = cBuf + (size_t)NG * NCH * 512;
  float* Hin  = cDee + (size_t)NG * NCH * 32;
  float* path = Hin  + (size_t)NG * NCH * 512; // 4*L

  fsb_front_wmma<<<dim3(L_ / 16, NG), 128, 0, stream>>>(x, perms, w_in, b_in, Win, xcB, zB);
  fsb_midmix<<<dim3(L_ / 256, NG), 256, 0, stream>>>(xcB, convw, convb, Wx, Wdt, bdt,
                                                     uB, dtB, BcB, CcB);
  fsb_scan1<<<dim3(NCH, NG), 512, 0, stream>>>(dtB, uB, BcB, Alog, cBuf, cDee);
  fsb_scan2<<<NG, 512, 0, stream>>>(cBuf, cDee, Alog, Hin);
  fsb_scan3<<<dim3(NCH, NG), 512, 0, stream>>>(dtB, uB, BcB, CcB, zB, Hin, Alog, Dp, yzB);
  fsb_back_wmma<<<dim3(L_ / 64, NG), 128, 0, stream>>>(yzB, Wout, ln_g, ln_b, w_out, b_out, path);
  fsb_gather<<<(L_ + 255) / 256, 256, 0, stream>>>(path, perms, out);
}